// LowPass_37211596653112
// MI455X (gfx1250) — compile-verified
//
#include <hip/hip_runtime.h>

// ---------------------------------------------------------------------------
// Lowpass biquad over x[32][262144] via chunked affine scan + WMMA tiles.
//   pass1: per-chunk zero-state run -> terminal states      (WMMA)
//   combine: 512-step affine scan with constant M^L         (1 wave, f64)
//   pass2: exact per-chunk run with scanned init -> output  (WMMA)
// ---------------------------------------------------------------------------

typedef float v8f __attribute__((ext_vector_type(8)));
typedef float v2f __attribute__((ext_vector_type(2)));

constexpr int T      = 262144;      // samples per channel
constexpr int L      = 512;         // chunk length
constexpr int NB     = T / L;       // 512 chunks per channel
constexpr int NTILE  = L / 16;      // 32 WMMA tiles per chunk
constexpr int CG     = 2;           // two groups of 16 channels
constexpr int UNITS  = CG * NB;     // 1024 wave-units
constexpr int WPG    = 8;           // waves per 256-thread workgroup
constexpr int GRID   = UNITS / WPG; // 128 workgroups

struct CoefBlock {
    float  b0, b1, b2, a1, a2;
    float  pad[3];
    double m00, m01, m10, m11;      // M^L (cross-chunk state transfer)
};

// ---------------------------------------------------------------------------
// Kernel 1: coefficients + M^L (one thread, double precision, runs once)
// ---------------------------------------------------------------------------
__global__ void biquad_coef_kernel(const float* __restrict__ fr,
                                   const float* __restrict__ qr,
                                   const int*   __restrict__ srp,
                                   CoefBlock*   __restrict__ cb) {
    const double MIN_F = 200.0, MAX_F = 18000.0, MIN_Q = 0.5, MAX_Q = 10.0;
    const double PI = 3.14159265358979323846;
    double f  = 1.0 / (1.0 + exp(-(double)fr[0])) * (MAX_F - MIN_F) + MIN_F;
    double Q  = 1.0 / (1.0 + exp(-(double)qr[0])) * (MAX_Q - MIN_Q) + MIN_Q;
    double sr = (double)srp[0];
    double w0 = 2.0 * PI * f / sr;
    double cw = cos(w0), sw = sin(w0);
    double alpha = sw / (2.0 * Q);
    double a0 = 1.0 + alpha;
    double b0 = ((1.0 - cw) * 0.5) / a0;
    double b1 = (1.0 - cw) / a0;
    double a1 = (-2.0 * cw) / a0;
    double a2 = (1.0 - alpha) / a0;
    cb->b0 = (float)b0; cb->b1 = (float)b1; cb->b2 = (float)b0;
    cb->a1 = (float)a1; cb->a2 = (float)a2;
    // M = [[-a1,-a2],[1,0]];  M^512 by 9 squarings
    double m00 = -a1, m01 = -a2, m10 = 1.0, m11 = 0.0;
    for (int i = 0; i < 9; ++i) {
        double t00 = m00*m00 + m01*m10, t01 = m00*m01 + m01*m11;
        double t10 = m10*m00 + m11*m10, t11 = m10*m01 + m11*m11;
        m00 = t00; m01 = t01; m10 = t10; m11 = t11;
    }
    cb->m00 = m00; cb->m01 = m01; cb->m10 = m10; cb->m11 = m11;
}

// ---------------------------------------------------------------------------
// Kernel 3: cross-chunk affine scan  s_b = M^L * s_{b-1} + t_b  (32 lanes)
// ---------------------------------------------------------------------------
__global__ void biquad_combine_kernel(const CoefBlock* __restrict__ cb,
                                      const float2*    __restrict__ term,
                                      float2*          __restrict__ init) {
    int lane = threadIdx.x;
    if (lane >= 32) return;
    int cg = lane >> 4, n = lane & 15;
    const double m00 = cb->m00, m01 = cb->m01, m10 = cb->m10, m11 = cb->m11;
    double s1 = 0.0, s2 = 0.0;                 // (y[-1], y[-2]) entering chunk b
    for (int b = 0; b < NB; ++b) {
        int idx = (cg * NB + b) * 16 + n;
        init[idx] = make_float2((float)s1, (float)s2);
        float2 t = term[idx];
        double n1 = m00 * s1 + m01 * s2 + (double)t.x;
        double n2 = m10 * s1 + m11 * s2 + (double)t.y;
        s1 = n1; s2 = n2;
    }
}

// ---------------------------------------------------------------------------
// Kernels 2 & 4: per-chunk WMMA run. WRITE=false -> record terminal state.
//                WRITE=true  -> use scanned init, write clamped output.
// ---------------------------------------------------------------------------
__device__ __forceinline__ float bperm_f(int srcLane, float v) {
    return __int_as_float(__builtin_amdgcn_ds_bpermute(srcLane << 2,
                                                       __float_as_int(v)));
}

template <bool WRITE>
__global__ __launch_bounds__(256) void biquad_pass_kernel(
    const float*  __restrict__ x,
    float*        __restrict__ out,
    const CoefBlock* __restrict__ cb,
    const float2* __restrict__ initArr,
    float2*       __restrict__ termArr)
{
    const int lane  = threadIdx.x & 31;
    const int wid   = blockIdx.x * WPG + (threadIdx.x >> 5);
    const int cg    = wid & 1;          // channel group (16 channels)
    const int chunk = wid >> 1;         // time chunk
    const int m     = lane & 15;        // row index within lane-half
    const int kh    = lane >> 4;        // lane half

    const float b0 = cb->b0, b1 = cb->b1, b2 = cb->b2;
    const float a1 = cb->a1, a2 = cb->a2;

    // --- IIR impulse response f[0..15] (register array, constant indices) ---
    float fv[16];
    fv[0] = 1.f; fv[1] = -a1;
#pragma unroll
    for (int k = 2; k < 16; ++k) fv[k] = -a1 * fv[k-1] - a2 * fv[k-2];

    // --- A fragments: A_k = H[:, 4k:4k+4], H[m][c] = (m>=c) ? f[m-c] : 0 ---
    // A(16x4 f32) layout: element (m,c) -> lane m + 16*(c>=2), VGPR c&1.
    v2f afr[4];
#pragma unroll
    for (int k = 0; k < 4; ++k) { afr[k].x = 0.f; afr[k].y = 0.f; }
#pragma unroll
    for (int j = 0; j < 16; ++j) {
#pragma unroll
        for (int k = 0; k < 4; ++k) {
            const int c0 = 4 * k;                  // + 2*kh at runtime
            afr[k].x = (m - 2*kh - c0     == j) ? fv[j] : afr[k].x;
            afr[k].y = (m - 2*kh - c0 - 1 == j) ? fv[j] : afr[k].y;
        }
    }

    // --- homogeneous responses G1 (y[-1]) and G2 (y[-2]) -------------------
    float Gv1[16], Gv2[16];
    Gv1[0] = -a1;               Gv2[0] = -a2;
    Gv1[1] = -a1*Gv1[0] - a2;   Gv2[1] = -a1*Gv2[0];
#pragma unroll
    for (int k = 2; k < 16; ++k) {
        Gv1[k] = -a1*Gv1[k-1] - a2*Gv1[k-2];
        Gv2[k] = -a1*Gv2[k-1] - a2*Gv2[k-2];
    }
    // C/D layout: VGPR r holds time m = r + 8*kh for this lane.
    float g1[8], g2[8];
#pragma unroll
    for (int r = 0; r < 8; ++r) {
        g1[r] = kh ? Gv1[r + 8] : Gv1[r];
        g2[r] = kh ? Gv2[r + 8] : Gv2[r];
    }

    const int    ch = cg * 16 + m;                 // this lane's channel (B/D: N = m)
    const float* xr = x + (size_t)ch * T;
    float*       orow = out + (size_t)ch * T;

    float y1 = 0.f, y2 = 0.f;                      // (y[-1], y[-2])
    if constexpr (WRITE) {
        float2 s = initArr[(cg * NB + chunk) * 16 + m];
        y1 = s.x; y2 = s.y;
    }

    const int t0 = chunk * L;
    const float* pp = xr + t0 + 2 * kh;            // per-lane staging pointer
    float*       op = orow + t0 + 8 * kh;          // per-lane store pointer

    // ---- stage tile 0 (guard x[-1], x[-2] only for chunk 0) ----
    float2 lo[4], hi[4];
    if (chunk == 0) {
#pragma unroll
        for (int k = 0; k < 4; ++k) {
            const int q0 = 4 * k + 2 * kh;
            lo[k].x = (q0 >= 2) ? xr[q0 - 2] : 0.f;
            lo[k].y = (q0 >= 1) ? xr[q0 - 1] : 0.f;
            hi[k]   = *reinterpret_cast<const float2*>(xr + q0);
        }
    } else {
#pragma unroll
        for (int k = 0; k < 4; ++k) {
            lo[k] = *reinterpret_cast<const float2*>(pp + 4 * k - 2);
            hi[k] = *reinterpret_cast<const float2*>(pp + 4 * k);
        }
    }

#pragma unroll 2
    for (int tile = 0; tile < NTILE; ++tile) {
        // B fragments from staged registers:
        //   v[t] = b0 x[t] + b1 x[t-1] + b2 x[t-2]
        v2f bfr[4];
#pragma unroll
        for (int k = 0; k < 4; ++k) {
            bfr[k].x = fmaf(b0, hi[k].x, fmaf(b1, lo[k].y, b2 * lo[k].x));
            bfr[k].y = fmaf(b0, hi[k].y, fmaf(b1, hi[k].x, b2 * lo[k].y));
        }

        // Prefetch next tile while the WMMA chain runs.
        if (tile + 1 < NTILE) {
            const float* pn = pp + 16 * (tile + 1);
#pragma unroll
            for (int k = 0; k < 4; ++k) {
                lo[k] = *reinterpret_cast<const float2*>(pn + 4 * k - 2);
                hi[k] = *reinterpret_cast<const float2*>(pn + 4 * k);
            }
        }

        // C := initial-state correction  C[m][n] = y1*G1[m] + y2*G2[m]
        v8f c;
#pragma unroll
        for (int r = 0; r < 8; ++r) c[r] = fmaf(y1, g1[r], y2 * g2[r]);

        // D = H * V + C   (chain of four 16x16x4 f32 WMMAs, D->C forwarded)
#pragma unroll
        for (int k = 0; k < 4; ++k)
            c = __builtin_amdgcn_wmma_f32_16x16x4_f32(
                    false, afr[k], false, bfr[k], (short)0, c, false, false);

        // New state: y[15] -> VGPR7 lanes 16..31, y[14] -> VGPR6 lanes 16..31.
        y1 = bperm_f(m + 16, c[7]);
        y2 = bperm_f(m + 16, c[6]);

        if constexpr (WRITE) {
            // Lane's 8 D values are 8 consecutive times of channel `ch`.
            float4 s0, s1v;
            s0.x  = fminf(fmaxf(c[0], -1.f), 1.f);
            s0.y  = fminf(fmaxf(c[1], -1.f), 1.f);
            s0.z  = fminf(fmaxf(c[2], -1.f), 1.f);
            s0.w  = fminf(fmaxf(c[3], -1.f), 1.f);
            s1v.x = fminf(fmaxf(c[4], -1.f), 1.f);
            s1v.y = fminf(fmaxf(c[5], -1.f), 1.f);
            s1v.z = fminf(fmaxf(c[6], -1.f), 1.f);
            s1v.w = fminf(fmaxf(c[7], -1.f), 1.f);
            float4* o4 = reinterpret_cast<float4*>(op + 16 * tile); // 16B aligned
            o4[0] = s0;
            o4[1] = s1v;
        }
    }

    if constexpr (!WRITE) {
        if (lane < 16)
            termArr[(cg * NB + chunk) * 16 + lane] = make_float2(y1, y2);
    }
}

// ---------------------------------------------------------------------------
extern "C" void kernel_launch(void* const* d_in, const int* in_sizes, int n_in,
                              void* d_out, int out_size, void* d_ws, size_t ws_size,
                              hipStream_t stream) {
    const float* x    = (const float*)d_in[0];
    const float* fr   = (const float*)d_in[1];
    const float* qr   = (const float*)d_in[2];
    const int*   sr   = (const int*)  d_in[3];
    float*       out  = (float*)d_out;

    CoefBlock* cb   = (CoefBlock*)d_ws;
    float2*    term = (float2*)((char*)d_ws + 64);
    float2*    init = term + (size_t)CG * NB * 16;   // 2*512*16 float2 each

    biquad_coef_kernel<<<1, 1, 0, stream>>>(fr, qr, sr, cb);
    biquad_pass_kernel<false><<<GRID, 256, 0, stream>>>(x, nullptr, cb, nullptr, term);
    biquad_combine_kernel<<<1, 32, 0, stream>>>(cb, term, init);
    biquad_pass_kernel<true><<<GRID, 256, 0, stream>>>(x, out, cb, init, nullptr);
}